// CandidatePenaltyCrossEntropyCriterion_55525337203267
// MI455X (gfx1250) — compile-verified
//
#include <hip/hip_runtime.h>
#include <stdint.h>

#define IGNORE_INDEX (-100)
#define TPB 512
#define NBUF 5          // LDS chunk buffers; prefetch depth = NBUF-1 = 4
#define CV4 512         // float4 vectors per chunk (one per thread)
#define CHUNK_F (CV4 * 4)

typedef float v2f __attribute__((ext_vector_type(2)));
typedef float v8f __attribute__((ext_vector_type(8)));

__device__ __forceinline__ unsigned lds_addr_of(const void* p) {
  // AS(3) generic pointers are {aperture_hi32, lds_offset32}.
  return (unsigned)(uintptr_t)p;
}

// Exact fp32 sum of 512 block partials using v_wmma_f32_16x16x4_f32.
// B is all-ones, so D[m,n] = sum_k A[m,k]; 8 accumulating WMMAs cover all 512
// partials exactly once; D rows collapse with 7 adds + 2 readlanes.
__device__ __forceinline__ float block_sum_wmma(float v, float* red, float* bc) {
  const int t = threadIdx.x;
  red[t] = v;
  __syncthreads();
  if (t < 32) {  // wave 0, EXEC all ones inside this branch
    v8f acc = {0.f, 0.f, 0.f, 0.f, 0.f, 0.f, 0.f, 0.f};
    v2f one;
    one[0] = 1.0f;
    one[1] = 1.0f;
    const int m16 = t & 15;
    const int k0 = (t >> 4) << 1;
#pragma unroll
    for (int it = 0; it < 8; ++it) {
      v2f a;
      a[0] = red[it * 64 + (k0 + 0) * 16 + m16];
      a[1] = red[it * 64 + (k0 + 1) * 16 + m16];
      acc = __builtin_amdgcn_wmma_f32_16x16x4_f32(false, a, false, one,
                                                  (short)0, acc, false, false);
    }
    float s = acc[0] + acc[1] + acc[2] + acc[3] + acc[4] + acc[5] + acc[6] + acc[7];
    float tot = __uint_as_float(__builtin_amdgcn_readlane(__float_as_uint(s), 0)) +
                __uint_as_float(__builtin_amdgcn_readlane(__float_as_uint(s), 16));
    if (t == 0) *bc = tot;
  }
  __syncthreads();
  return *bc;
}

__device__ __forceinline__ void wait_async_le(int n) {
  switch (n) {
    case 0: asm volatile("s_wait_asynccnt 0x0" ::: "memory"); break;
    case 1: asm volatile("s_wait_asynccnt 0x1" ::: "memory"); break;
    case 2: asm volatile("s_wait_asynccnt 0x2" ::: "memory"); break;
    default: asm volatile("s_wait_asynccnt 0x3" ::: "memory"); break;
  }
}

// Issue one async b128 HBM->LDS transfer per lane for chunk c. Address is
// clamped so every wave issues every chunk (keeps per-wave ASYNCcnt uniform).
__device__ __forceinline__ void issue_chunk(const float* gbase, float* buf,
                                            int c, int t, int N4) {
  int v = c * CV4 + t;
  if (v > N4 - 1) v = N4 - 1;  // clamp: redundant load, distinct LDS slot
  unsigned lds = lds_addr_of(&buf[(c % NBUF) * CHUNK_F + 4 * t]);
  unsigned long long ga = (unsigned long long)(uintptr_t)(gbase + 4 * (size_t)v);
  asm volatile("global_load_async_to_lds_b128 %0, %1, off"
               :: "v"(lds), "v"(ga) : "memory");
}

// Per (b,j): first-occurrence flag; also zeroes the loss accumulator.
__global__ void __launch_bounds__(TPB)
precompute_kernel(const int* __restrict__ target, int* __restrict__ flags,
                  float* __restrict__ acc, int S) {
  __shared__ int tg[TPB];
  const int b = blockIdx.x;
  const int j = threadIdx.x;
  tg[j] = target[b * S + j];
  __syncthreads();
  const int tj = tg[j];
  int f = 1;
  for (int k = 0; k < j; ++k) {
    if (tg[k] == tj) { f = 0; break; }
  }
  flags[b * S + j] = f;
  if (b == 0 && j == 0) acc[0] = 0.0f;
}

// One block per token row: barrier-free async-pipelined streaming + per-thread
// online softmax, then WMMA block reductions, then candidate penalties.
__global__ void __launch_bounds__(TPB)
row_loss_kernel(const float* __restrict__ logits,
                const int* __restrict__ target,
                const int* __restrict__ firstflag,
                float* __restrict__ acc,
                int S, int V) {
  __shared__ float buf[NBUF * CHUNK_F];  // 40 KB
  __shared__ float red[TPB];
  __shared__ float bcM, bcZ, bcT;

  const int t = threadIdx.x;
  const int r = blockIdx.x;  // r = b*S + i
  const int b = r / S;
  const int i = r - b * S;
  const float* rowp = logits + (size_t)r * (size_t)V;

  // Alignment peel: rows are only 4B-aligned.
  const uintptr_t baddr = (uintptr_t)rowp;
  const int a0 = (int)(((16u - (unsigned)(baddr & 15u)) & 15u) >> 2);  // 0..3
  const int N4 = (V - a0) >> 2;            // aligned float4 count
  const int NC = (N4 + CV4 - 1) / CV4;     // chunks
  const float* gbase = rowp + a0;          // 16B aligned

  // Thread-local online softmax state.
  float m = -__builtin_inff();
  float s = 0.0f;

  // Head/tail scalars handled by thread 0 straight from global.
  if (t == 0) {
    for (int e = 0; e < a0; ++e) {
      float x = rowp[e];
      if (x > m) { s *= __expf(m - x); m = x; }
      s += __expf(x - m);
    }
    for (int e = a0 + 4 * N4; e < V; ++e) {
      float x = rowp[e];
      if (x > m) { s *= __expf(m - x); m = x; }
      s += __expf(x - m);
    }
  }

  // Prologue: fill the pipeline (chunks 0 .. NBUF-2).
  for (int c = 0; c < NBUF - 1 && c < NC; ++c) issue_chunk(gbase, buf, c, t, N4);

  // Barrier-free main loop: each wave consumes only LDS bytes its own lanes
  // wrote, so per-wave ASYNCcnt waits are the only synchronization needed.
  for (int c = 0; c < NC; ++c) {
    int need = NC - 1 - c;
    if (need > NBUF - 2) need = NBUF - 2;
    wait_async_le(need);  // chunk c complete for this wave (in-order)

    const int v = c * CV4 + t;
    if (v < N4) {
      const float4 x = *(const float4*)&buf[(c % NBUF) * CHUNK_F + 4 * t];
      const float m4 = fmaxf(fmaxf(x.x, x.y), fmaxf(x.z, x.w));
      if (m4 > m) { s *= __expf(m - m4); m = m4; }
      s += __expf(x.x - m) + __expf(x.y - m) + __expf(x.z - m) + __expf(x.w - m);
    }

    const int cn = c + NBUF - 1;
    if (cn < NC) {
      // Ensure our ds_load of the oldest slot drained before DMA reuses it.
      asm volatile("s_wait_dscnt 0x0" ::: "memory");
      issue_chunk(gbase, buf, cn, t, N4);
    }
  }

  // Combine (m,s) pairs: block max, rescale, WMMA-sum.
  red[t] = m;
  __syncthreads();
  for (int sN = TPB / 2; sN > 0; sN >>= 1) {
    if (t < sN) red[t] = fmaxf(red[t], red[t + sN]);
    __syncthreads();
  }
  if (t == 0) bcM = red[0];
  __syncthreads();
  const float M = bcM;

  const float sp = s * __expf(m - M);  // m==-inf => s==0 => contributes 0
  const float Z = block_sum_wmma(sp, red, &bcZ);
  const float lse = M + __logf(Z);

  // Candidate penalties (thread t == prior position j) + MLE term (thread i).
  const int* trow = target + b * S;
  float part = 0.0f;
  const int tgt_i = trow[i];
  if (t < i) {
    const int tj = trow[t];
    if (tj != IGNORE_INDEX && tj != tgt_i && firstflag[b * S + t]) {
      const float p = __expf(rowp[tj] - lse);  // L2-resident gather
      float omp = 1.0f - p;
      omp = (omp < 1e-5f) ? 1e-5f : omp;
      part += -__logf(omp);
    }
  }
  if (t == i && tgt_i != IGNORE_INDEX) {
    part += -(rowp[tgt_i] - lse);
  }
  const float tot = block_sum_wmma(part, red, &bcT);
  if (t == 0) atomicAdd(acc, tot);
}

__global__ void __launch_bounds__(256)
finalize_kernel(const int* __restrict__ target, const float* __restrict__ acc,
                float* __restrict__ out, int n) {
  __shared__ int cnt[256];
  const int t = threadIdx.x;
  int c = 0;
  for (int idx = t; idx < n; idx += 256) c += (target[idx] != IGNORE_INDEX) ? 1 : 0;
  cnt[t] = c;
  __syncthreads();
  for (int sN = 128; sN > 0; sN >>= 1) {
    if (t < sN) cnt[t] += cnt[t + sN];
    __syncthreads();
  }
  if (t == 0) out[0] = acc[0] / (float)cnt[0];
}

extern "C" void kernel_launch(void* const* d_in, const int* in_sizes, int n_in,
                              void* d_out, int out_size, void* d_ws, size_t ws_size,
                              hipStream_t stream) {
  const float* logits = (const float*)d_in[0];
  const int* target = (const int*)d_in[1];

  const int BS = in_sizes[1];      // B*S = 1024
  const int S = 512;               // per reference
  const int B = BS / S;            // 2
  const int V = in_sizes[0] / BS;  // 50257

  float* acc = (float*)d_ws;                  // [0]: loss accumulator
  int* flags = (int*)((char*)d_ws + 16);      // [BS]: first-occurrence flags

  precompute_kernel<<<B, S, 0, stream>>>(target, flags, acc, S);
  row_loss_kernel<<<BS, TPB, 0, stream>>>(logits, target, flags, acc, S, V);
  finalize_kernel<<<1, 256, 0, stream>>>(target, acc, (float*)d_out, BS);
}